// MPN_40879498728983
// MI455X (gfx1250) — compile-verified
//
#include <hip/hip_runtime.h>
#include <hip/hip_bf16.h>

typedef __attribute__((ext_vector_type(2))) float v2f;
typedef __attribute__((ext_vector_type(8))) float v8f;

#define NN 50000
#define NE 800000
#define DIN 64
#define DHID 128
#define DOUT 64

// ---------------------------------------------------------------------------
// Edge-parallel scatter-add: agg[dst[e], :] += x[src[e], :]
// One thread handles 4 consecutive floats of one edge's row (coalesced float4
// gather; 4x global_atomic_add_f32 scatter, L2-resident).
// ---------------------------------------------------------------------------
__global__ __launch_bounds__(256) void scatter_add_kernel(
    const float* __restrict__ x, const int* __restrict__ src,
    const int* __restrict__ dst, float* __restrict__ agg, int E, int D)
{
    int tid = blockIdx.x * blockDim.x + threadIdx.x;
    int chunks = D >> 2;                 // float4 chunks per row
    int e = tid / chunks;
    if (e >= E) return;
    int c = (tid - e * chunks) << 2;
    int s = src[e];
    int d = dst[e];
    float4 v = *(const float4*)(x + (size_t)s * D + c);
    float* o = agg + (size_t)d * D + c;
    atomicAdd(o + 0, v.x);
    atomicAdd(o + 1, v.y);
    atomicAdd(o + 2, v.z);
    atomicAdd(o + 3, v.w);
}

#define WMMA_F32(a, b, c) \
    __builtin_amdgcn_wmma_f32_16x16x4_f32(false, (a), false, (b), (short)0, (c), false, false)

// ---------------------------------------------------------------------------
// OUT[M,N] = act( A[M,K] @ Wa[K,N] + X[M,K] @ Wx[K,N] + bias )
// fp32 WMMA (V_WMMA_F32_16X16X4_F32), register-blocked: one wave owns a
// 32x32 output tile = 2x2 accumulators. Each A fragment feeds 2 N-tiles and
// each B fragment feeds 2 M-tiles -> 12 VMEM per 8 WMMA (vs 3:1 unblocked).
// Tail rows (M % 32 != 0): load rows clamp to M-1 (valid memory, results
// discarded), stores row-guarded. All divergence is after the WMMAs, so EXEC
// is all-ones around every matrix op.
// ---------------------------------------------------------------------------
template <int RELU, int K, int N>
__global__ __launch_bounds__(128) void gemm_dual_kernel(
    const float* __restrict__ A,    // [M,K]  aggregated neighbors
    const float* __restrict__ X,    // [M,K]  root features
    const float* __restrict__ Wa,   // [K,N]
    const float* __restrict__ Wx,   // [K,N]
    const float* __restrict__ bias, // [N]
    float* __restrict__ out,        // [M,N]
    int M)
{
    const int lane = threadIdx.x & 31;
    const int wave = threadIdx.x >> 5;
    const int tilesN = N >> 5;                       // 32-wide column blocks
    const int tilesM = (M + 31) >> 5;                // 32-tall row blocks
    const int tile = blockIdx.x * 4 + wave;          // 4 waves / block
    if (tile >= tilesM * tilesN) return;             // wave-uniform

    const int tm = tile / tilesN;
    const int tn = tile - tm * tilesN;
    const int r0 = tm << 5;

    // A-fragment (16x4): row = lane&15; VGPRs hold K {0,1} (lanes 0-15) or
    // {2,3} (lanes 16-31). B-fragment (4x16): col across lanes, same K split.
    const int mA = r0 + (lane & 15);
    const int m0 = mA < M ? mA : M - 1;              // clamp (tail tile only)
    const int m1 = (mA + 16) < M ? (mA + 16) : M - 1;
    const int n0 = (tn << 5) + (lane & 15);
    const int n1 = n0 + 16;
    const int kOff = (lane >> 4) << 1;               // 0 or 2

    const float* aR0 = A + (size_t)m0 * K;
    const float* aR1 = A + (size_t)m1 * K;
    const float* xR0 = X + (size_t)m0 * K;
    const float* xR1 = X + (size_t)m1 * K;

    v8f c00 = {}, c01 = {}, c10 = {}, c11 = {};
#pragma unroll 2
    for (int k0 = 0; k0 < K; k0 += 4) {
        const int kk = k0 + kOff;
        v2f a0, a1, x0, x1, wa0, wa1, wx0, wx1;
        a0.x = aR0[kk];  a0.y = aR0[kk + 1];         // contiguous -> b64
        a1.x = aR1[kk];  a1.y = aR1[kk + 1];
        x0.x = xR0[kk];  x0.y = xR0[kk + 1];
        x1.x = xR1[kk];  x1.y = xR1[kk + 1];
        wa0.x = Wa[(size_t)kk * N + n0];  wa0.y = Wa[(size_t)(kk + 1) * N + n0];
        wa1.x = Wa[(size_t)kk * N + n1];  wa1.y = Wa[(size_t)(kk + 1) * N + n1];
        wx0.x = Wx[(size_t)kk * N + n0];  wx0.y = Wx[(size_t)(kk + 1) * N + n0];
        wx1.x = Wx[(size_t)kk * N + n1];  wx1.y = Wx[(size_t)(kk + 1) * N + n1];

        c00 = WMMA_F32(a0, wa0, c00);  c00 = WMMA_F32(x0, wx0, c00);
        c01 = WMMA_F32(a0, wa1, c01);  c01 = WMMA_F32(x0, wx1, c01);
        c10 = WMMA_F32(a1, wa0, c10);  c10 = WMMA_F32(x1, wx0, c10);
        c11 = WMMA_F32(a1, wa1, c11);  c11 = WMMA_F32(x1, wx1, c11);
    }

    // C/D layout: VGPR v at lane l holds (row = v + (l>=16)*8, col = l&15)
    const float bv0 = bias[n0];
    const float bv1 = bias[n1];
    const int rOff = (lane >> 4) << 3;
#pragma unroll
    for (int v = 0; v < 8; ++v) {
        const int row0 = r0 + v + rOff;              // M-subtile 0
        const int row1 = row0 + 16;                  // M-subtile 1
        float r00 = c00[v] + bv0, r01 = c01[v] + bv1;
        float r10 = c10[v] + bv0, r11 = c11[v] + bv1;
        if (RELU) {
            r00 = fmaxf(r00, 0.0f); r01 = fmaxf(r01, 0.0f);
            r10 = fmaxf(r10, 0.0f); r11 = fmaxf(r11, 0.0f);
        }
        if (row0 < M) {
            out[(size_t)row0 * N + n0] = r00;
            out[(size_t)row0 * N + n1] = r01;
        }
        if (row1 < M) {
            out[(size_t)row1 * N + n0] = r10;
            out[(size_t)row1 * N + n1] = r11;
        }
    }
}

// ---------------------------------------------------------------------------
extern "C" void kernel_launch(void* const* d_in, const int* in_sizes, int n_in,
                              void* d_out, int out_size, void* d_ws, size_t ws_size,
                              hipStream_t stream)
{
    const float* x      = (const float*)d_in[0];
    const int*   eidx   = (const int*)d_in[1];   // [2, NE]
    const float* wrel1  = (const float*)d_in[2];
    const float* wroot1 = (const float*)d_in[3];
    const float* b1     = (const float*)d_in[4];
    const float* wrel2  = (const float*)d_in[5];
    const float* wroot2 = (const float*)d_in[6];
    const float* b2     = (const float*)d_in[7];
    const float* wrel3  = (const float*)d_in[8];
    const float* wroot3 = (const float*)d_in[9];
    const float* b3     = (const float*)d_in[10];

    const int* src = eidx;
    const int* dst = eidx + NE;

    float* h1  = (float*)d_ws;                       // [NN, 128]
    float* agg = h1  + (size_t)NN * DHID;            // [NN, 128] (reused at width 64)
    float* h2  = agg + (size_t)NN * DHID;            // [NN, 128]

    // ---------------- layer 1 (64 -> 128, relu) ----------------
    hipMemsetAsync(agg, 0, (size_t)NN * DIN * sizeof(float), stream);
    {
        long long threads = (long long)NE * (DIN / 4);
        scatter_add_kernel<<<(int)((threads + 255) / 256), 256, 0, stream>>>(
            x, src, dst, agg, NE, DIN);
    }
    {
        int tiles = ((NN + 31) / 32) * (DHID / 32);  // 1563 * 4 = 6252 waves
        gemm_dual_kernel<1, DIN, DHID><<<(tiles + 3) / 4, 128, 0, stream>>>(
            agg, x, wrel1, wroot1, b1, h1, NN);
    }

    // ---------------- layer 2 (128 -> 128, relu) ----------------
    hipMemsetAsync(agg, 0, (size_t)NN * DHID * sizeof(float), stream);
    {
        long long threads = (long long)NE * (DHID / 4);
        scatter_add_kernel<<<(int)((threads + 255) / 256), 256, 0, stream>>>(
            h1, src, dst, agg, NE, DHID);
    }
    {
        int tiles = ((NN + 31) / 32) * (DHID / 32);
        gemm_dual_kernel<1, DHID, DHID><<<(tiles + 3) / 4, 128, 0, stream>>>(
            agg, h1, wrel2, wroot2, b2, h2, NN);
    }

    // ---------------- layer 3 (128 -> 64, no relu) ----------------
    hipMemsetAsync(agg, 0, (size_t)NN * DHID * sizeof(float), stream);
    {
        long long threads = (long long)NE * (DHID / 4);
        scatter_add_kernel<<<(int)((threads + 255) / 256), 256, 0, stream>>>(
            h2, src, dst, agg, NE, DHID);
    }
    {
        int tiles = ((NN + 31) / 32) * (DOUT / 32);  // 1563 * 2 = 3126 waves
        gemm_dual_kernel<0, DHID, DOUT><<<(tiles + 3) / 4, 128, 0, stream>>>(
            agg, h2, wrel3, wroot3, b3, (float*)d_out, NN);
    }
}